// MultiMechanismFusion_18957985644693
// MI455X (gfx1250) — compile-verified
//
#include <hip/hip_runtime.h>
#include <hip/hip_bf16.h>
#include <math.h>

// Problem constants (match reference)
#define B_   32
#define D_   128
#define T_   1024
#define H_   256
#define KK_  3
#define OUT_ 2

typedef __attribute__((ext_vector_type(2))) float v2f;
typedef __attribute__((ext_vector_type(8))) float v8f;

__device__ __forceinline__ v8f wmma_f32(v2f a, v2f b, v8f c) {
  // D = A(16x4,f32) * B(4x16,f32) + C(16x16,f32)  -> v_wmma_f32_16x16x4_f32
  return __builtin_amdgcn_wmma_f32_16x16x4_f32(false, a, false, b, (short)0, c,
                                               false, false);
}

__device__ __forceinline__ v2f ld2(const float* p) {
  v2f r; r[0] = p[0]; r[1] = p[1]; return r;
}

__device__ __forceinline__ v8f v8zero() {
  v8f z = {0.f, 0.f, 0.f, 0.f, 0.f, 0.f, 0.f, 0.f};
  return z;
}

// ---------------------------------------------------------------------------
// prep: [B,D,1,T] -> [B,T,D] transpose + per-(b,t) squared norms
// ---------------------------------------------------------------------------
__global__ void prep_kernel(const float* __restrict__ in, float* __restrict__ xbtd,
                            float* __restrict__ sq) {
  int idx = blockIdx.x * 256 + threadIdx.x;          // over B*T
  if (idx >= B_ * T_) return;
  int b = idx >> 10, t = idx & (T_ - 1);
  const float* p = in + (size_t)b * D_ * T_ + t;     // stride T over d
  float* o = xbtd + (size_t)idx * D_;
  float s = 0.f;
#pragma unroll 8
  for (int d = 0; d < D_; ++d) {
    float v = p[(size_t)d * T_];
    o[d] = v;
    s += v * v;
  }
  sq[idx] = s;
}

// mean over T of in[b][d][t] (original [B,D,1,T] layout, t contiguous)
__global__ void colmean_kernel(const float* __restrict__ in, float* __restrict__ mean) {
  int idx = blockIdx.x * 256 + threadIdx.x;          // over B*D
  if (idx >= B_ * D_) return;
  const float* p = in + (size_t)idx * T_;
  float s = 0.f;
#pragma unroll 8
  for (int t = 0; t < T_; ++t) s += p[t];
  mean[idx] = s * (1.f / (float)T_);
}

// gate[b,d] = sigmoid(mean[b,:] @ W[:,d] + b[d])
__global__ void gate_kernel(const float* __restrict__ mean, const float* __restrict__ W,
                            const float* __restrict__ bias, float* __restrict__ gate) {
  int idx = blockIdx.x * 256 + threadIdx.x;          // over B*D
  if (idx >= B_ * D_) return;
  int b = idx >> 7, d = idx & (D_ - 1);
  float acc = bias[d];
  const float* m = mean + b * D_;
#pragma unroll 8
  for (int k = 0; k < D_; ++k) acc += m[k] * W[k * D_ + d];
  gate[idx] = 1.f / (1.f + expf(-acc));
}

// ---------------------------------------------------------------------------
// Projection GEMM: C[M,128] = X[M,128] @ W[128,128] + bias   (M = B*T)
// block = 256 threads (8 waves), tile = 128 rows x 64 cols
// ---------------------------------------------------------------------------
#define GA_S 36
__global__ void gemm_proj_kernel(const float* __restrict__ X, const float* __restrict__ W,
                                 const float* __restrict__ bias, float* __restrict__ C) {
  __shared__ float Al[128 * GA_S];   // A rows x 32 k (padded)
  __shared__ float Bt[64 * GA_S];    // W^T: 64 n x 32 k (padded)
  int tid = threadIdx.x, lane = tid & 31, wv = tid >> 5;
  int m = lane & 15, hf = lane >> 4;
  int rb = blockIdx.x * 128;
  int nb = blockIdx.y * 64;

  v8f acc[4] = {v8zero(), v8zero(), v8zero(), v8zero()};

  for (int kc = 0; kc < 4; ++kc) {                    // 128 / 32 chunks
    int k0 = kc * 32;
    __syncthreads();
    for (int i = tid; i < 128 * 32; i += 256) {
      int r = i >> 5, k = i & 31;
      Al[r * GA_S + k] = X[(size_t)(rb + r) * D_ + k0 + k];
    }
    for (int i = tid; i < 64 * 32; i += 256) {
      int n = i >> 5, k = i & 31;
      Bt[n * GA_S + k] = W[(size_t)(k0 + k) * D_ + nb + n];
    }
    __syncthreads();
#pragma unroll
    for (int ks = 0; ks < 8; ++ks) {
      v2f a = ld2(&Al[(wv * 16 + m) * GA_S + ks * 4 + 2 * hf]);
#pragma unroll
      for (int nt = 0; nt < 4; ++nt) {
        v2f b = ld2(&Bt[(nt * 16 + m) * GA_S + ks * 4 + 2 * hf]);
        acc[nt] = wmma_f32(a, b, acc[nt]);
      }
    }
  }
#pragma unroll
  for (int nt = 0; nt < 4; ++nt) {
    float bv = bias[nb + nt * 16 + m];
#pragma unroll
    for (int i = 0; i < 8; ++i) {
      int row = rb + wv * 16 + i + 8 * hf;
      C[(size_t)row * D_ + nb + nt * 16 + m] = acc[nt][i] + bv;
    }
  }
}

// ---------------------------------------------------------------------------
// Flash cross-attention: O = softmax(Q K^T / sqrt(D)) V   per batch
// block = 128 threads (4 waves), 64 query rows per block
// ---------------------------------------------------------------------------
#define AQ_S 132
#define AP_S 18
__global__ void attn_kernel(const float* __restrict__ Q, const float* __restrict__ K,
                            const float* __restrict__ V, float* __restrict__ O) {
  __shared__ float Qs[64 * AQ_S];
  __shared__ float Ksl[16 * AQ_S];
  __shared__ float Vsl[16 * AQ_S];
  __shared__ float Ps[4 * 16 * AP_S];
  const float scale = 0.088388347648318447f;   // 1/sqrt(128)

  int b = blockIdx.y;
  int qb = blockIdx.x * 64;
  int tid = threadIdx.x, lane = tid & 31, wv = tid >> 5;
  int m = lane & 15, hf = lane >> 4;

  const float* Qg = Q + (size_t)(b * T_ + qb) * D_;
  for (int i = tid; i < 64 * D_; i += 128) {
    int r = i >> 7, d = i & (D_ - 1);
    Qs[r * AQ_S + d] = Qg[(size_t)r * D_ + d];
  }

  v8f acc[8];
  float mrow[8], lrow[8];
#pragma unroll
  for (int i = 0; i < 8; ++i) { acc[i] = v8zero(); mrow[i] = -1e30f; lrow[i] = 0.f; }

  for (int jb = 0; jb < T_ / 16; ++jb) {
    __syncthreads();
    const float* Kg = K + (size_t)(b * T_ + jb * 16) * D_;
    const float* Vg = V + (size_t)(b * T_ + jb * 16) * D_;
    for (int i = tid; i < 16 * D_; i += 128) {
      int r = i >> 7, d = i & (D_ - 1);
      Ksl[r * AQ_S + d] = Kg[(size_t)r * D_ + d];
      Vsl[r * AQ_S + d] = Vg[(size_t)r * D_ + d];
    }
    __syncthreads();

    // S tile = Q(16x128) * K^T(128x16)
    v8f s = v8zero();
#pragma unroll
    for (int ks = 0; ks < 32; ++ks) {
      v2f a = ld2(&Qs[(wv * 16 + m) * AQ_S + ks * 4 + 2 * hf]);
      v2f bb = ld2(&Ksl[m * AQ_S + ks * 4 + 2 * hf]);    // K^T[k][j=m] = K[m][k]
      s = wmma_f32(a, bb, s);
    }

    // online softmax, stage P in per-wave LDS patch
#pragma unroll
    for (int i = 0; i < 8; ++i) {
      float sv = s[i] * scale;
      float rm = sv;
      rm = fmaxf(rm, __shfl_xor(rm, 1));
      rm = fmaxf(rm, __shfl_xor(rm, 2));
      rm = fmaxf(rm, __shfl_xor(rm, 4));
      rm = fmaxf(rm, __shfl_xor(rm, 8));
      float nm = fmaxf(mrow[i], rm);
      float p = expf(sv - nm);
      float rs = p;
      rs += __shfl_xor(rs, 1);
      rs += __shfl_xor(rs, 2);
      rs += __shfl_xor(rs, 4);
      rs += __shfl_xor(rs, 8);
      float fac = expf(mrow[i] - nm);
      lrow[i] = lrow[i] * fac + rs;
      mrow[i] = nm;
#pragma unroll
      for (int dt = 0; dt < 8; ++dt) acc[dt][i] *= fac;
      Ps[wv * 16 * AP_S + (i + 8 * hf) * AP_S + m] = p;   // C-layout -> LDS
    }

    // O += P(16x16) * V(16x128)
#pragma unroll
    for (int ks2 = 0; ks2 < 4; ++ks2) {
      int k0 = ks2 * 4 + 2 * hf;
      v2f a = ld2(&Ps[wv * 16 * AP_S + m * AP_S + k0]);   // A-layout read-back
#pragma unroll
      for (int dt = 0; dt < 8; ++dt) {
        v2f bb;
        bb[0] = Vsl[k0 * AQ_S + dt * 16 + m];
        bb[1] = Vsl[(k0 + 1) * AQ_S + dt * 16 + m];
        acc[dt] = wmma_f32(a, bb, acc[dt]);
      }
    }
  }

#pragma unroll
  for (int i = 0; i < 8; ++i) {
    float inv = 1.f / lrow[i];
    int row = b * T_ + qb + wv * 16 + i + 8 * hf;
#pragma unroll
    for (int dt = 0; dt < 8; ++dt)
      O[(size_t)row * D_ + dt * 16 + m] = acc[dt][i] * inv;
  }
}

// ---------------------------------------------------------------------------
// fuse: build concat tensor xcat[B, 3D, T]
// ---------------------------------------------------------------------------
__global__ void fuse_kernel(const float* __restrict__ O1, const float* __restrict__ O2,
                            const float* __restrict__ Xf, const float* __restrict__ Xs,
                            const float* __restrict__ gatef, const float* __restrict__ gates,
                            const float* __restrict__ aw, float* __restrict__ xcat) {
  size_t idx = (size_t)blockIdx.x * 256 + threadIdx.x;  // B*3D*T
  if (idx >= (size_t)B_ * 3 * D_ * T_) return;
  int t = (int)(idx % T_);
  int c = (int)((idx / T_) % (3 * D_));
  int b = (int)(idx / ((size_t)3 * D_ * T_));
  size_t base = ((size_t)(b * T_ + t)) * D_;
  float val;
  if (c < D_) {
    val = O1[base + c] + O2[base + c];
  } else if (c < 2 * D_) {
    int d = c - D_;
    val = gatef[b * D_ + d] * Xf[base + d] + gates[b * D_ + d] * Xs[base + d];
  } else {
    int d = c - 2 * D_;
    float al = 1.f / (1.f + expf(-aw[0]));
    val = al * Xf[base + d] + (1.f - al) * Xs[base + d];
  }
  xcat[idx] = val;
}

// ---------------------------------------------------------------------------
// Causal dilated conv1d + ReLU as 3 shifted WMMA GEMMs.
// in [B,Cin,T], w [H,Cin,K], out [B,H,T].  block = 128 thr (4 waves),
// tile = 64 h x 64 t, Cin chunked by 32 with 8-column left halo.
// ---------------------------------------------------------------------------
#define CW_S 34
#define CX_S 74
__global__ void conv_kernel(const float* __restrict__ X, const float* __restrict__ Wt,
                            const float* __restrict__ bias, float* __restrict__ Y,
                            int Cin, int dil) {
  __shared__ float Wl[3 * 64 * CW_S];
  __shared__ float Xl[32 * CX_S];
  int b = blockIdx.z;
  int hb = blockIdx.y * 64;
  int tb = blockIdx.x * 64;
  int tid = threadIdx.x, lane = tid & 31, wv = tid >> 5;
  int m = lane & 15, hf = lane >> 4;

  v8f acc[4] = {v8zero(), v8zero(), v8zero(), v8zero()};
  int nch = Cin >> 5;

  for (int cc = 0; cc < nch; ++cc) {
    __syncthreads();
    for (int i = tid; i < 3 * 64 * 32; i += 128) {
      int k = i / 2048;
      int rem = i - k * 2048;
      int hl = rem >> 5, cl = rem & 31;
      Wl[(k * 64 + hl) * CW_S + cl] =
          Wt[((size_t)(hb + hl) * Cin + cc * 32 + cl) * KK_ + k];
    }
    for (int i = tid; i < 32 * 72; i += 128) {
      int cl = i / 72, tt = i - cl * 72;
      int tg = tb - 8 + tt;
      Xl[cl * CX_S + tt] =
          (tg >= 0) ? X[((size_t)b * Cin + cc * 32 + cl) * T_ + tg] : 0.f;
    }
    __syncthreads();
#pragma unroll
    for (int k = 0; k < KK_; ++k) {
      int off = 8 - (KK_ - 1 - k) * dil;
#pragma unroll
      for (int ks = 0; ks < 8; ++ks) {
        v2f a = ld2(&Wl[(k * 64 + wv * 16 + m) * CW_S + ks * 4 + 2 * hf]);
        int c0 = ks * 4 + 2 * hf;
#pragma unroll
        for (int nt = 0; nt < 4; ++nt) {
          int col = nt * 16 + m + off;
          v2f bb;
          bb[0] = Xl[c0 * CX_S + col];
          bb[1] = Xl[(c0 + 1) * CX_S + col];
          acc[nt] = wmma_f32(a, bb, acc[nt]);
        }
      }
    }
  }
#pragma unroll
  for (int i = 0; i < 8; ++i) {
    int h = hb + wv * 16 + i + 8 * hf;
    float bv = bias[h];
#pragma unroll
    for (int nt = 0; nt < 4; ++nt) {
      float v = fmaxf(acc[nt][i] + bv, 0.f);
      Y[((size_t)b * H_ + h) * T_ + tb + nt * 16 + m] = v;
    }
  }
}

// mean over T of y[b][h][t]
__global__ void meant_kernel(const float* __restrict__ Y, float* __restrict__ ym) {
  int idx = blockIdx.x * 256 + threadIdx.x;    // B*H
  if (idx >= B_ * H_) return;
  const float* p = Y + (size_t)idx * T_;
  float s = 0.f;
#pragma unroll 8
  for (int t = 0; t < T_; ++t) s += p[t];
  ym[idx] = s * (1.f / (float)T_);
}

__global__ void head_kernel(const float* __restrict__ ym, const float* __restrict__ fcw,
                            const float* __restrict__ fcb, float* __restrict__ out) {
  int tid = threadIdx.x;                        // 64 = B*OUT
  if (tid >= B_ * OUT_) return;
  int b = tid >> 1, o = tid & 1;
  float s = fcb[o];
#pragma unroll 8
  for (int h = 0; h < H_; ++h) s += ym[b * H_ + h] * fcw[h * OUT_ + o];
  out[b * OUT_ + o] = s;
}

// ---------------------------------------------------------------------------
// MMD: per-t 32x32 Grams via WMMA; block per t, 4 waves = 2x2 tile grid
// ---------------------------------------------------------------------------
#define MM_S 130
__global__ void mmd_kernel(const float* __restrict__ Xf, const float* __restrict__ Xs,
                           const float* __restrict__ sqf, const float* __restrict__ sqs,
                           float* __restrict__ partial) {
  __shared__ float Fl[32 * MM_S];
  __shared__ float Sl[32 * MM_S];
  __shared__ float sfl[32], ssl[32], red[4];
  const float cc = 0.5f;                        // 1/(2*sigma^2)

  int t = blockIdx.x;
  int tid = threadIdx.x, lane = tid & 31, wv = tid >> 5;
  int m = lane & 15, hf = lane >> 4;
  for (int i = tid; i < 32 * D_; i += 128) {
    int bi = i >> 7, d = i & (D_ - 1);
    Fl[bi * MM_S + d] = Xf[((size_t)(bi * T_ + t)) * D_ + d];
    Sl[bi * MM_S + d] = Xs[((size_t)(bi * T_ + t)) * D_ + d];
  }
  if (tid < 32) { sfl[tid] = sqf[tid * T_ + t]; ssl[tid] = sqs[tid * T_ + t]; }
  __syncthreads();

  int rt = wv >> 1, ct = wv & 1;
  v8f dxx = v8zero(), dyy = v8zero(), dxy = v8zero();
#pragma unroll
  for (int ks = 0; ks < 32; ++ks) {
    int k0 = ks * 4 + 2 * hf;
    v2f af = ld2(&Fl[(rt * 16 + m) * MM_S + k0]);
    v2f as = ld2(&Sl[(rt * 16 + m) * MM_S + k0]);
    v2f bf = ld2(&Fl[(ct * 16 + m) * MM_S + k0]);
    v2f bs = ld2(&Sl[(ct * 16 + m) * MM_S + k0]);
    dxx = wmma_f32(af, bf, dxx);
    dyy = wmma_f32(as, bs, dyy);
    dxy = wmma_f32(af, bs, dxy);
  }
  float local = 0.f;
#pragma unroll
  for (int i = 0; i < 8; ++i) {
    int bi = rt * 16 + i + 8 * hf;
    int bj = ct * 16 + m;
    float kxx = expf(-cc * (sfl[bi] + sfl[bj] - 2.f * dxx[i]));
    float kyy = expf(-cc * (ssl[bi] + ssl[bj] - 2.f * dyy[i]));
    float kxy = expf(-cc * (sfl[bi] + ssl[bj] - 2.f * dxy[i]));
    local += kxx + kyy - 2.f * kxy;
  }
  local += __shfl_xor(local, 16);
  local += __shfl_xor(local, 8);
  local += __shfl_xor(local, 4);
  local += __shfl_xor(local, 2);
  local += __shfl_xor(local, 1);
  if (lane == 0) red[wv] = local;
  __syncthreads();
  if (tid == 0) partial[t] = red[0] + red[1] + red[2] + red[3];
}

__global__ void mmd_final_kernel(const float* __restrict__ partial, float* __restrict__ out) {
  float s = 0.f;
  for (int i = 0; i < T_; ++i) s += partial[i];
  out[B_ * OUT_] = s * (1.f / ((float)B_ * (float)B_ * (float)T_));
}

// ---------------------------------------------------------------------------
extern "C" void kernel_launch(void* const* d_in, const int* in_sizes, int n_in,
                              void* d_out, int out_size, void* d_ws, size_t ws_size,
                              hipStream_t stream) {
  (void)in_sizes; (void)out_size; (void)ws_size;
  const float* fast   = (const float*)d_in[0];
  const float* slow   = (const float*)d_in[1];
  const float* caf_qw = (const float*)d_in[2];
  const float* caf_qb = (const float*)d_in[3];
  const float* caf_kw = (const float*)d_in[4];
  const float* caf_kb = (const float*)d_in[5];
  const float* caf_vw = (const float*)d_in[6];
  const float* caf_vb = (const float*)d_in[7];
  const float* cas_qw = (const float*)d_in[8];
  const float* cas_qb = (const float*)d_in[9];
  const float* cas_kw = (const float*)d_in[10];
  const float* cas_kb = (const float*)d_in[11];
  const float* cas_vw = (const float*)d_in[12];
  const float* cas_vb = (const float*)d_in[13];
  const float* gf_w   = (const float*)d_in[14];
  const float* gf_b   = (const float*)d_in[15];
  const float* gs_w   = (const float*)d_in[16];
  const float* gs_b   = (const float*)d_in[17];
  const float* aw     = (const float*)d_in[18];

  const float *tw0, *tw1, *tw2, *tb0, *tb1, *tb2, *fcw, *fcb;
  if (n_in >= 27) {            // tuples flattened to separate inputs
    tw0 = (const float*)d_in[19]; tw1 = (const float*)d_in[20]; tw2 = (const float*)d_in[21];
    tb0 = (const float*)d_in[22]; tb1 = (const float*)d_in[23]; tb2 = (const float*)d_in[24];
    fcw = (const float*)d_in[25]; fcb = (const float*)d_in[26];
  } else {                     // tuples concatenated into single inputs
    const float* tws = (const float*)d_in[19];
    tw0 = tws; tw1 = tws + (size_t)H_ * 3 * D_ * KK_; tw2 = tw1 + (size_t)H_ * H_ * KK_;
    const float* tbs = (const float*)d_in[20];
    tb0 = tbs; tb1 = tbs + H_; tb2 = tbs + 2 * H_;
    fcw = (const float*)d_in[21]; fcb = (const float*)d_in[22];
  }

  // workspace layout (floats)
  const size_t BTD = (size_t)B_ * T_ * D_;   // 4,194,304
  const size_t BT  = (size_t)B_ * T_;
  const size_t BD  = (size_t)B_ * D_;
  float* w = (float*)d_ws;
  float* Xf    = w;
  float* Xs    = Xf + BTD;
  float* sqf   = Xs + BTD;
  float* sqs   = sqf + BT;
  float* meanf = sqs + BT;
  float* means = meanf + BD;
  float* gatef = means + BD;
  float* gates = gatef + BD;
  float* Qf    = gates + BD;
  float* Kf    = Qf + BTD;
  float* Vf    = Kf + BTD;
  float* Qs2   = Vf + BTD;
  float* Ks2   = Qs2 + BTD;
  float* Vs2   = Ks2 + BTD;
  float* O1    = Vs2 + BTD;
  float* O2    = O1 + BTD;
  float* ymean = O2 + BTD;                 // B*H
  float* mpart = ymean + (size_t)B_ * H_;  // T floats
  // aliased regions (stream-ordered reuse after producers are dead):
  float* xcat  = Qf;     // 3*BTD, built after attention consumed Q/K/V
  float* Y1    = Qs2;    // 2*BTD
  float* Y2    = O1;     // 2*BTD
  float* Y3    = Qf;     // 2*BTD, after xcat consumed by conv1

  float* out = (float*)d_out;

  prep_kernel<<<(int)((BT + 255) / 256), 256, 0, stream>>>(fast, Xf, sqf);
  prep_kernel<<<(int)((BT + 255) / 256), 256, 0, stream>>>(slow, Xs, sqs);
  colmean_kernel<<<(int)((BD + 255) / 256), 256, 0, stream>>>(fast, meanf);
  colmean_kernel<<<(int)((BD + 255) / 256), 256, 0, stream>>>(slow, means);
  gate_kernel<<<(int)((BD + 255) / 256), 256, 0, stream>>>(meanf, gf_w, gf_b, gatef);
  gate_kernel<<<(int)((BD + 255) / 256), 256, 0, stream>>>(means, gs_w, gs_b, gates);

  dim3 gproj(B_ * T_ / 128, D_ / 64);
  gemm_proj_kernel<<<gproj, 256, 0, stream>>>(Xf, caf_qw, caf_qb, Qf);
  gemm_proj_kernel<<<gproj, 256, 0, stream>>>(Xs, caf_kw, caf_kb, Kf);
  gemm_proj_kernel<<<gproj, 256, 0, stream>>>(Xs, caf_vw, caf_vb, Vf);
  gemm_proj_kernel<<<gproj, 256, 0, stream>>>(Xs, cas_qw, cas_qb, Qs2);
  gemm_proj_kernel<<<gproj, 256, 0, stream>>>(Xf, cas_kw, cas_kb, Ks2);
  gemm_proj_kernel<<<gproj, 256, 0, stream>>>(Xf, cas_vw, cas_vb, Vs2);

  dim3 gattn(T_ / 64, B_);
  attn_kernel<<<gattn, 128, 0, stream>>>(Qf, Kf, Vf, O1);
  attn_kernel<<<gattn, 128, 0, stream>>>(Qs2, Ks2, Vs2, O2);

  size_t nfuse = (size_t)B_ * 3 * D_ * T_;
  fuse_kernel<<<(int)((nfuse + 255) / 256), 256, 0, stream>>>(O1, O2, Xf, Xs, gatef,
                                                              gates, aw, xcat);

  dim3 gconv(T_ / 64, H_ / 64, B_);
  conv_kernel<<<gconv, 128, 0, stream>>>(xcat, tw0, tb0, Y1, 3 * D_, 1);
  conv_kernel<<<gconv, 128, 0, stream>>>(Y1,   tw1, tb1, Y2, H_,     2);
  conv_kernel<<<gconv, 128, 0, stream>>>(Y2,   tw2, tb2, Y3, H_,     4);

  meant_kernel<<<(B_ * H_ + 255) / 256, 256, 0, stream>>>(Y3, ymean);
  head_kernel<<<1, 64, 0, stream>>>(ymean, fcw, fcb, out);

  mmd_kernel<<<T_, 128, 0, stream>>>(Xf, Xs, sqf, sqs, mpart);
  mmd_final_kernel<<<1, 1, 0, stream>>>(mpart, out);
}